// OriginalConjunctionLayer_20495583937170
// MI455X (gfx1250) — compile-verified
//
#include <hip/hip_runtime.h>

typedef float v2f __attribute__((ext_vector_type(2)));
typedef float v8f __attribute__((ext_vector_type(8)));

#define B_DIM 128
#define D_DIM 2048
#define N_DIM 512
#define P_TERMS 20   // Mercator terms: tail < 0.5^20/21 per element (uw < 0.5)

// One wave computes one 16x16 output tile via V_WMMA_F32_16X16X4_F32.
// Block: 256 threads = 8 waves; one 16-row M strip, 8 adjacent 16-col N tiles.
__global__ __launch_bounds__(256)
void conjunction_wmma_kernel(const float* __restrict__ X,   // [128 x 2048]
                             const float* __restrict__ W,   // [2048 x 512]
                             float* __restrict__ Out) {     // [128 x 512]
  extern __shared__ float xs[];              // x strip transposed: xs[d*16 + m]
  const int tid  = threadIdx.x;
  const int wave = tid >> 5;
  const int lane = tid & 31;
  const int mt = blockIdx.x & 7;             // 8 M tiles
  const int ng = blockIdx.x >> 3;            // 4 N groups of 8 tiles
  const int b0 = mt * 16;
  const int n0 = (ng * 8 + wave) * 16;

  // ---- Stage X[b0:b0+16, :] (contiguous 32768 floats) into LDS, transposed.
  const float* Xblk = X + b0 * D_DIM;
  for (int idx = tid; idx < 16 * D_DIM; idx += 256) {
    float v = Xblk[idx];
    int m = idx >> 11;            // idx / 2048
    int d = idx & (D_DIM - 1);
    xs[d * 16 + m] = v;
  }
  __syncthreads();

  // ---- WMMA lane mapping (ISA 7.12.2):
  // A 16x4 f32: lanes 0-15 -> M=lane, VGPR{0,1}=K{0,1}; lanes 16-31 -> K{2,3}.
  // B 4x16 f32: lanes 0-15 -> N=lane, VGPR{0,1}=K{0,1}; lanes 16-31 -> K{2,3}.
  const int mn   = lane & 15;                // A row / B col within tile
  const int kofs = (lane < 16) ? 0 : 2;      // K pair selector

  v8f accs[4];                               // series accumulators (break C-chain)
  v8f accb = {0,0,0,0,0,0,0,0};              // binarized-path accumulator
#pragma unroll
  for (int i = 0; i < 4; ++i) accs[i] = (v8f){0,0,0,0,0,0,0,0};

  const float* Wc = W + n0 + mn;             // column base for this lane

  for (int d0 = 0; d0 < D_DIM; d0 += 4) {
    const int dk = d0 + kofs;
    // A operands from LDS (conflict-free: 16 consecutive floats per half-wave)
    float x0 = xs[dk * 16 + mn];
    float x1 = xs[(dk + 1) * 16 + mn];
    // B operands from global (64B contiguous per half-wave; L2 resident)
    float w0 = Wc[(size_t)dk * N_DIM];
    float w1 = Wc[(size_t)(dk + 1) * N_DIM];

    if (d0 + 20 < D_DIM)                     // pull next W rows into cache
      __builtin_prefetch(Wc + (size_t)(dk + 16) * N_DIM, 0, 0);

    float u0 = 1.0f - x0, u1 = 1.0f - x1;

    // ---- Binarized hard-AND path: exp( GEMM(log x, (w > 0.5)) )
    float lx0 = __logf(fmaxf(x0, 1e-30f));
    float lx1 = __logf(fmaxf(x1, 1e-30f));
    float bw0 = (w0 > 0.5f) ? 1.0f : 0.0f;
    float bw1 = (w1 > 0.5f) ? 1.0f : 0.0f;
    v2f a, b;
    a[0] = lx0; a[1] = lx1;
    b[0] = bw0; b[1] = bw1;
    accb = __builtin_amdgcn_wmma_f32_16x16x4_f32(
        false, a, false, b, (short)0, accb, false, false);

    // ---- Continuous path: -sum_d log(1-u*w) = sum_k (1/k) GEMM(u^k, w^k)
    float p0 = u0, p1 = u1, q0 = w0, q1 = w1;
#pragma unroll
    for (int k = 1; k <= P_TERMS; ++k) {
      const float rk = 1.0f / (float)k;      // folds to literal after unroll
      a[0] = p0 * rk; a[1] = p1 * rk;
      b[0] = q0;      b[1] = q1;
      accs[k & 3] = __builtin_amdgcn_wmma_f32_16x16x4_f32(
          false, a, false, b, (short)0, accs[k & 3], false, false);
      p0 *= u0; p1 *= u1; q0 *= w0; q1 *= w1;
    }
  }

  // ---- Epilogue: C/D 16x16 f32 layout: VGPR j -> M=j (lanes 0-15) / M=j+8.
  const int mrow = (lane >> 4) << 3;
  const int ncol = n0 + (lane & 15);
#pragma unroll
  for (int j = 0; j < 8; ++j) {
    float S = accs[0][j] + accs[1][j] + accs[2][j] + accs[3][j]; // = -sum(log soft)
    float cont = 1.0f / (1.0f + S);          // -1/(-1 + sum(log soft))
    float binr = __expf(accb[j]);            // == 1 when no weight > 0.5
    float outv = (binr - cont) + cont;       // reference's graft arithmetic
    Out[(size_t)(b0 + j + mrow) * N_DIM + ncol] = outv;
  }
}

extern "C" void kernel_launch(void* const* d_in, const int* in_sizes, int n_in,
                              void* d_out, int out_size, void* d_ws, size_t ws_size,
                              hipStream_t stream) {
  const float* X = (const float*)d_in[0];    // inputs  [128 x 2048] f32
  const float* W = (const float*)d_in[1];    // weights [2048 x 512] f32
  float* Out = (float*)d_out;                // [128 x 512] f32
  (void)in_sizes; (void)n_in; (void)out_size; (void)d_ws; (void)ws_size;

  dim3 grid(32);                              // 8 M tiles x 4 N groups
  dim3 block(256);                            // 8 waves; 1 wave = 16x16 tile
  size_t shmem = (size_t)D_DIM * 16 * sizeof(float);  // 128 KB x-strip
  conjunction_wmma_kernel<<<grid, block, shmem, stream>>>(X, W, Out);
}